// _fasterRCNN_85529978732841
// MI455X (gfx1250) — compile-verified
//
#include <hip/hip_runtime.h>
#include <stdint.h>

#define PRE      300
#define POSTK    100
#define THREADS  1024
#define SELPAD   512
#define CAP      3072                 /* candidate-list capacity (expected ~1K) */
#define NWORD    ((PRE + 31) / 32)    /* 10 */
#define NMS_T    0.3f
#define SCORE_T  0.05f

#if defined(__HIP_DEVICE_COMPILE__) && defined(__gfx1250__)
#define DEV_ASM 1
#else
#define DEV_ASM 0
#endif

/* phase-pool: candidates (24KB)  ->  box/delta staging (9.6KB)  ->  supMat (12KB) */
#define POOL_BYTES (CAP * 4 * 2)

__device__ __forceinline__ unsigned score_key(float s) {
  // scores are >= 0; order-preserving map to uint, 0 reserved for "invalid"
  if (!(s >= SCORE_T)) return 0u;
  return __float_as_uint(s) | 0x80000000u;
}

// Block-parallel crossing search over an LDS histogram.
// Each thread owns 4 bins (uint4 load), inclusive suffix-scan over thread
// partials, then the unique thread containing the crossing writes results.
// All threads must call (contains barriers).
__device__ __forceinline__ void select_level(const unsigned* __restrict__ hist,
                                             int nbins, unsigned base,
                                             unsigned* __restrict__ scan,
                                             unsigned* s_bin, unsigned* s_above,
                                             int tid) {
  const int ngroups = nbins >> 2;
  uint4 h4 = make_uint4(0u, 0u, 0u, 0u);
  unsigned s = 0u;
  if (tid < ngroups) {
    h4 = ((const uint4*)hist)[tid];
    s = h4.x + h4.y + h4.z + h4.w;
  }
  scan[tid] = s;
  __syncthreads();
  // inclusive suffix scan: scan[t] = sum_{t' >= t} s[t']
  for (int d = 1; d < THREADS; d <<= 1) {
    unsigned v = (tid + d < THREADS) ? scan[tid + d] : 0u;
    __syncthreads();
    scan[tid] += v;
    __syncthreads();
  }
  const unsigned incl = scan[tid];
  const unsigned excl = incl - s;
  if (tid < ngroups && (base + excl) < (unsigned)PRE && (base + incl) >= (unsigned)PRE) {
    unsigned run = base + excl;
    unsigned hh[4] = {h4.x, h4.y, h4.z, h4.w};
#pragma unroll
    for (int j = 3; j >= 0; --j) {
      unsigned c = hh[j];
      if (run + c >= (unsigned)PRE) { *s_bin = (unsigned)(tid * 4 + j); *s_above = run; break; }
      run += c;
    }
  }
  __syncthreads();
}

__global__ __launch_bounds__(THREADS)
void proposal_nms_kernel(const float* __restrict__ cls_prob,
                         const float* __restrict__ boxes,
                         const float* __restrict__ deltas,
                         const float* __restrict__ im_info,
                         float* __restrict__ out,
                         int nbox) {
  const int b   = blockIdx.x;
  const int tid = threadIdx.x;

  __shared__ __align__(16) unsigned hist[4096];
  __shared__ unsigned            scanArr[THREADS];
  __shared__ unsigned long long  sel[SELPAD];
  __shared__ unsigned            tieIdx[SELPAD];
  __shared__ __align__(16) unsigned char pool[POOL_BYTES];
  __shared__ float               X1[PRE], Y1[PRE], X2[PRE], Y2[PRE], AR[PRE];
  __shared__ unsigned char       VALID[PRE];
  __shared__ int                 outIdx[POSTK];
  __shared__ unsigned            s_bin, s_above, s_gcnt, s_tcnt, s_ccnt;

  /* phase-pool views (temporally disjoint, separated by barriers) */
  unsigned* candK = (unsigned*)pool;                           /* [CAP]      */
  unsigned* candI = (unsigned*)(pool + CAP * 4);               /* [CAP]      */
  float (*bx)[4] = (float(*)[4])pool;                          /* [PRE][4]   */
  float (*dx)[4] = (float(*)[4])(pool + PRE * 16);             /* [PRE][4]   */
  unsigned (*supMat)[NWORD] = (unsigned(*)[NWORD])pool;        /* [PRE][10]  */

  const float4* sc4 = (const float4*)(cls_prob + (size_t)b * 2u * (size_t)nbox);
  const int npair = nbox >> 1;   /* one float4 = 2 (bg,fg) score pairs */

  // ===== pass 1 (HBM): level-1 histogram on key bits [31:20] =====
  for (int p = tid; p < 4096; p += THREADS) hist[p] = 0u;
  __syncthreads();
  for (int v = tid; v < npair; v += THREADS) {
    __builtin_prefetch(sc4 + v + THREADS, 0, 3);   // global_prefetch_b8
    float4 p = sc4[v];
    unsigned k0 = score_key(p.y);
    unsigned k1 = score_key(p.w);
    atomicAdd(&hist[k0 >> 20], 1u);
    atomicAdd(&hist[k1 >> 20], 1u);
  }
  __syncthreads();
  select_level(hist, 4096, 0u, scanArr, &s_bin, &s_above, tid);
  const unsigned b1 = s_bin;
  const unsigned above1 = s_above;
  __syncthreads();

  // ===== pass 2 (L2): collect winners (>b1) + candidates (==b1) + level-2 hist =====
  for (int p = tid; p < 4096; p += THREADS) hist[p] = 0u;
  for (int p = tid; p < SELPAD; p += THREADS) sel[p] = 0ull;
  if (tid == 0) { s_gcnt = 0u; s_tcnt = 0u; s_ccnt = 0u; }
  __syncthreads();
  for (int v = tid; v < npair; v += THREADS) {
    float4 p = sc4[v];
    unsigned kk[2] = { score_key(p.y), score_key(p.w) };
#pragma unroll
    for (int e = 0; e < 2; ++e) {
      unsigned k = kk[e];
      unsigned i = (unsigned)(2 * v + e);
      unsigned t12 = k >> 20;
      if (t12 > b1) {           // unconditionally in top-300 (count == above1 <= 299)
        unsigned pp = atomicAdd(&s_gcnt, 1u);
        sel[pp] = ((unsigned long long)k << 32) | (unsigned long long)(0xFFFFFFFFu - i);
      } else if (t12 == b1) {   // candidate: cache in LDS for levels 3 + gather
        unsigned q = atomicAdd(&s_ccnt, 1u);
        if (q < CAP) { candK[q] = k; candI[q] = i; }
        atomicAdd(&hist[(k >> 8) & 0xFFFu], 1u);
      }
    }
  }
  __syncthreads();
  select_level(hist, 4096, above1, scanArr, &s_bin, &s_above, tid);
  const unsigned b2 = s_bin;
  const unsigned above2 = s_above;
  const int cn = (int)(s_ccnt < (unsigned)CAP ? s_ccnt : (unsigned)CAP);
  __syncthreads();

  // ===== level 3: bits [7:0] over LDS candidate list only =====
  if (tid < 256) hist[tid] = 0u;
  __syncthreads();
  for (int q = tid; q < cn; q += THREADS) {
    unsigned k = candK[q];
    if (((k >> 8) & 0xFFFu) == b2) atomicAdd(&hist[k & 0xFFu], 1u);
  }
  __syncthreads();
  select_level(hist, 256, above2, scanArr, &s_bin, &s_above, tid);
  const unsigned b3 = s_bin;
  const unsigned above3 = s_above;               // count strictly greater than T, <= 299
  const unsigned T = (b1 << 20) | (b2 << 8) | b3;
  __syncthreads();

  // ===== gather remaining winners + ties from candidate list =====
  for (int q = tid; q < cn; q += THREADS) {
    unsigned k = candK[q];
    unsigned i = candI[q];
    if (k > T) {
      unsigned pp = atomicAdd(&s_gcnt, 1u);
      sel[pp] = ((unsigned long long)k << 32) | (unsigned long long)(0xFFFFFFFFu - i);
    } else if (k == T) {
      unsigned t = atomicAdd(&s_tcnt, 1u);
      if (t < SELPAD) tieIdx[t] = i;
    }
  }
  __syncthreads();

  // ties at T: take 'rem' smallest indices (top_k stability); typically rem == 1
  if (tid == 0) {
    unsigned rem = (unsigned)PRE - above3;
    unsigned tn = s_tcnt < (unsigned)SELPAD ? s_tcnt : (unsigned)SELPAD;
    for (unsigned j = 0; j < tn; ++j) hist[j] = 0u;   // hist reused as "used" flags
    for (unsigned r = 0; r < rem; ++r) {
      unsigned best = 0xFFFFFFFFu; int bj = -1;
      for (unsigned j = 0; j < tn; ++j)
        if (!hist[j] && tieIdx[j] < best) { best = tieIdx[j]; bj = (int)j; }
      if (bj < 0) break;
      hist[bj] = 1u;
      sel[above3 + r] = ((unsigned long long)T << 32) | (unsigned long long)(0xFFFFFFFFu - best);
    }
  }
  __syncthreads();

  // ===== bitonic sort descending, 512 slots (pads are 0 -> sink to end) =====
  for (int k = 2; k <= SELPAD; k <<= 1) {
    for (int j = k >> 1; j > 0; j >>= 1) {
      __syncthreads();
      if (tid < SELPAD) {
        int ixj = tid ^ j;
        if (ixj > tid) {
          unsigned long long a = sel[tid], c = sel[ixj];
          bool sw = ((tid & k) == 0) ? (a < c) : (a > c);
          if (sw) { sel[tid] = c; sel[ixj] = a; }
        }
      }
    }
  }
  __syncthreads();

  // ===== gather box + delta data for the 300 winners via async-to-LDS =====
  if (tid < PRE) {
    unsigned long long s = sel[tid];
    unsigned key = (unsigned)(s >> 32);
    unsigned idx = 0xFFFFFFFFu - (unsigned)(s & 0xFFFFFFFFull);
    bool v = (key != 0u) && (idx < (unsigned)nbox);
    VALID[tid] = v ? 1 : 0;
    if (!v) idx = 0u;
    const float* gb = boxes  + ((size_t)b * (size_t)nbox + idx) * 4u;
    const float* gd = deltas + ((size_t)b * (size_t)nbox + idx) * 4u;
#if DEV_ASM
    unsigned lds_b = (unsigned)(uintptr_t)(&bx[tid][0]);
    unsigned lds_d = (unsigned)(uintptr_t)(&dx[tid][0]);
    asm volatile("global_load_async_to_lds_b128 %0, %1, off"
                 :: "v"(lds_b), "v"(gb) : "memory");
    asm volatile("global_load_async_to_lds_b128 %0, %1, off"
                 :: "v"(lds_d), "v"(gd) : "memory");
#else
    float4 t0 = *(const float4*)gb;
    bx[tid][0] = t0.x; bx[tid][1] = t0.y; bx[tid][2] = t0.z; bx[tid][3] = t0.w;
    float4 t1 = *(const float4*)gd;
    dx[tid][0] = t1.x; dx[tid][1] = t1.y; dx[tid][2] = t1.z; dx[tid][3] = t1.w;
#endif
  }
#if DEV_ASM
#if __has_builtin(__builtin_amdgcn_s_wait_asynccnt)
  __builtin_amdgcn_s_wait_asynccnt(0);
#else
  asm volatile("s_wait_asynccnt 0x0" ::: "memory");
#endif
#endif
  __syncthreads();

  // ===== decode + clip only the 300 selected boxes =====
  if (tid < PRE) {
    float x1 = bx[tid][0], y1 = bx[tid][1], x2 = bx[tid][2], y2 = bx[tid][3];
    float w  = x2 - x1 + 1.0f;
    float h  = y2 - y1 + 1.0f;
    float cx = x1 + 0.5f * w;
    float cy = y1 + 0.5f * h;
    float pcx = (dx[tid][0] * 0.1f) * w + cx;
    float pcy = (dx[tid][1] * 0.1f) * h + cy;
    float pw  = expf(dx[tid][2] * 0.2f) * w;
    float ph  = expf(dx[tid][3] * 0.2f) * h;
    float hmax = im_info[b * 3 + 0] - 1.0f;
    float wmax = im_info[b * 3 + 1] - 1.0f;
    float nx1 = fminf(fmaxf(pcx - 0.5f * pw, 0.0f), wmax);
    float ny1 = fminf(fmaxf(pcy - 0.5f * ph, 0.0f), hmax);
    float nx2 = fminf(fmaxf(pcx + 0.5f * pw, 0.0f), wmax);
    float ny2 = fminf(fmaxf(pcy + 0.5f * ph, 0.0f), hmax);
    X1[tid] = nx1; Y1[tid] = ny1; X2[tid] = nx2; Y2[tid] = ny2;
    AR[tid] = (nx2 - nx1 + 1.0f) * (ny2 - ny1 + 1.0f);
  }
  __syncthreads();   /* also retires bx/dx before supMat overlays the pool */

  // ===== parallel 300x300 IoU suppression bitmask (div-free compare) =====
  for (int t = tid; t < PRE * NWORD; t += THREADS) {
    int i = t / NWORD;
    int w = t - i * NWORD;
    unsigned m = 0u;
    if (VALID[i]) {
      float xi1 = X1[i], yi1 = Y1[i], xi2 = X2[i], yi2 = Y2[i], ai = AR[i];
      int j0 = w * 32;
      int jend = j0 + 32 < PRE ? j0 + 32 : PRE;
      int jbeg = j0 > i + 1 ? j0 : i + 1;
      for (int j = jbeg; j < jend; ++j) {
        float ix1 = fmaxf(xi1, X1[j]);
        float iy1 = fmaxf(yi1, Y1[j]);
        float ix2 = fminf(xi2, X2[j]);
        float iy2 = fminf(yi2, Y2[j]);
        float iw = fmaxf(ix2 - ix1 + 1.0f, 0.0f);
        float ih = fmaxf(iy2 - iy1 + 1.0f, 0.0f);
        float inter = iw * ih;
        if (inter * (1.0f + NMS_T) > NMS_T * (ai + AR[j]))
          m |= 1u << (j - j0);
      }
    }
    supMat[i][w] = m;
  }
  __syncthreads();

  // ===== serial greedy reduction (exact reference semantics) + ranking =====
  if (tid == 0) {
    unsigned rm[NWORD];
#pragma unroll
    for (int w = 0; w < NWORD; ++w) rm[w] = 0u;
    for (int i = 0; i < PRE; ++i)
      if (!VALID[i]) rm[i >> 5] |= 1u << (i & 31);
    for (int i = 0; i < PRE; ++i) {
      if (!((rm[i >> 5] >> (i & 31)) & 1u)) {
#pragma unroll
        for (int w = 0; w < NWORD; ++w) rm[w] |= supMat[i][w];
      }
    }
    for (int k = 0; k < POSTK; ++k) outIdx[k] = -1;
    int r = 0;
    for (int i = 0; i < PRE && r < POSTK; ++i)
      if (!((rm[i >> 5] >> (i & 31)) & 1u)) outIdx[r++] = i;
  }
  __syncthreads();

  // ===== emit: (b, x1, y1, x2, y2) x 100, zero rows for missing =====
  if (tid < POSTK) {
    float* o = out + ((size_t)b * POSTK + tid) * 5u;
    int i = outIdx[tid];
    o[0] = (float)b;
    if (i >= 0) { o[1] = X1[i]; o[2] = Y1[i]; o[3] = X2[i]; o[4] = Y2[i]; }
    else        { o[1] = 0.0f;  o[2] = 0.0f;  o[3] = 0.0f;  o[4] = 0.0f; }
  }
}

extern "C" void kernel_launch(void* const* d_in, const int* in_sizes, int n_in,
                              void* d_out, int out_size, void* d_ws, size_t ws_size,
                              hipStream_t stream) {
  const float* cls   = (const float*)d_in[0];   // (B, N, 2)
  const float* boxes = (const float*)d_in[1];   // (B, N, 4)
  const float* delt  = (const float*)d_in[2];   // (B, N, 4)
  const float* imin  = (const float*)d_in[3];   // (B, 3)
  float* out = (float*)d_out;                   // (B, 100, 5) f32

  const int B = in_sizes[3] / 3;
  const int N = (B > 0) ? (in_sizes[0] / (2 * B)) : 0;

  proposal_nms_kernel<<<dim3((unsigned)B), dim3(THREADS), 0, stream>>>(
      cls, boxes, delt, imin, out, N);

  (void)n_in; (void)out_size; (void)d_ws; (void)ws_size;
}